// PatchMLSLAttention_38568806318651
// MI455X (gfx1250) — compile-verified
//
#include <hip/hip_runtime.h>

// ---------------- problem constants ----------------
#define B_   64
#define N_   4096
#define D_   256
#define CLS  20          // classes (padded to 32 for WMMA)
#define CH   8           // N-chunks per batch (parallel flash partials)
#define NG   8           // groups per chunk
#define GR   64          // rows per group (4 x 16-row WMMA tiles)

// ---------------- LDS layout (floats), rows padded to dodge bank conflicts ----
#define CBS  260                         // codebook row stride (256+4)
#define ES   260                         // E-tile row stride
#define PS   68                          // P row stride (64+4)

#define CB_OFF   0
#define E0_OFF   (32*CBS)                // 8320
#define E1_OFF   (E0_OFF + GR*ES)       // 24960
#define P_OFF    (E1_OFF + GR*ES)       // 41600
#define M_OFF    (P_OFF + 32*PS)        // 43776
#define L_OFF    (M_OFF + 32)
#define SC_OFF   (L_OFF + 32)
#define MN_OFF   (SC_OFF + 32)
#define TMAX_OFF (MN_OFF + 32)
#define TSUM_OFF (TMAX_OFF + 32*4)
#define LDS_FLOATS (TSUM_OFF + 32*4)    // 44160 floats = 176640 B (<320KB WGP)

typedef float v2f __attribute__((ext_vector_type(2)));
typedef float v8f __attribute__((ext_vector_type(8)));

static __device__ __forceinline__ v8f wmma_f32_k4(v2f a, v2f b, v8f c) {
  // D(16x16,f32) = A(16x4,f32) * B(4x16,f32) + C   -> v_wmma_f32_16x16x4_f32
  return __builtin_amdgcn_wmma_f32_16x16x4_f32(false, a, false, b, (short)0, c,
                                               false, false);
}

// Single-instruction xor-shuffle within 32 lanes: ds_swizzle_b32, group-of-32
// mode, offset = (xor_mask<<10) | (or_mask<<5) | and_mask(0x1F).
// xor masks < 16 never cross the 16-lane halves, so these reduce each half
// (the two M-rows of a WMMA C tile) independently.
#define SWZ_XOR(v, m) \
  __int_as_float(__builtin_amdgcn_ds_swizzle(__float_as_int(v), (0x001F | ((m) << 10))))

static __device__ __forceinline__ float half_reduce_max(float v) {
  v = fmaxf(v, SWZ_XOR(v, 1));
  v = fmaxf(v, SWZ_XOR(v, 2));
  v = fmaxf(v, SWZ_XOR(v, 4));
  v = fmaxf(v, SWZ_XOR(v, 8));
  return v;
}
static __device__ __forceinline__ float half_reduce_sum(float v) {
  v += SWZ_XOR(v, 1);
  v += SWZ_XOR(v, 2);
  v += SWZ_XOR(v, 4);
  v += SWZ_XOR(v, 8);
  return v;
}

// Async DMA of one 64x256 fp32 tile (row-major, rows padded to ES in LDS).
// 256 threads x 16 issues x 16B = 64KB. Tracked on ASYNCcnt.
static __device__ __forceinline__ void async_tile_load(const float* gRow0,
                                                       float* ldsRow0) {
  const int tid = threadIdx.x;
#pragma unroll
  for (int j = 0; j < 16; ++j) {
    const int id  = j * 256 + tid;     // consecutive tids -> consecutive 16B
    const int row = id >> 6;           // 64 x 16B per row
    const int cp  = id & 63;
    // low 32 bits of a flat LDS pointer are the wave-relative LDS byte offset
    unsigned lds = (unsigned)(size_t)(ldsRow0 + row * ES + cp * 4);
    unsigned long long ga = (unsigned long long)(size_t)(gRow0 + (size_t)row * D_ + cp * 4);
    asm volatile("global_load_async_to_lds_b128 %0, %1, off"
                 :: "v"(lds), "v"(ga) : "memory");
  }
}

__global__ __launch_bounds__(256, 1)
void flash_partial(const float* __restrict__ pe, const float* __restrict__ cbG,
                   float* __restrict__ pAcc, float* __restrict__ pM,
                   float* __restrict__ pL) {
  extern __shared__ float sm[];
  float* cb   = sm + CB_OFF;
  float* Pm   = sm + P_OFF;
  float* m_s  = sm + M_OFF;
  float* l_s  = sm + L_OFF;
  float* sc_s = sm + SC_OFF;
  float* mn_s = sm + MN_OFF;
  float* tmax = sm + TMAX_OFF;
  float* tsum = sm + TSUM_OFF;

  const int tid  = threadIdx.x;
  const int lane = tid & 31;
  const int wv   = tid >> 5;   // 8 waves
  const int t    = wv & 3;     // n-tile (0..3) in group   (S-phase role)
  const int mb   = wv >> 2;    // class block (0..1)       (S-phase role)
  const int hi   = lane >> 4;  // half-wave
  const int l16  = lane & 15;

  const int bq = blockIdx.x;   // b*CH + q
  const int b  = bq / CH;
  const int q  = bq % CH;
  const float* Eg = pe + ((size_t)b * N_ + (size_t)q * (NG * GR)) * D_;

  // codebook -> LDS (rows 20..31 zero)
  for (int i = tid; i < CLS * D_; i += 256)
    cb[(i >> 8) * CBS + (i & 255)] = cbG[i];
  for (int i = tid; i < (32 - CLS) * D_; i += 256)
    cb[(CLS + (i >> 8)) * CBS + (i & 255)] = 0.0f;
  if (tid < 32) { m_s[tid] = -3.0e38f; l_s[tid] = 0.0f; }

  const v8f vz = {0.f,0.f,0.f,0.f,0.f,0.f,0.f,0.f};
  v8f acc00 = vz, acc01 = vz, acc10 = vz, acc11 = vz;  // (mb x dtile), D-slice = wv*32

  async_tile_load(Eg, sm + E0_OFF);     // prefetch group 0
  __syncthreads();                      // also covers codebook/stat init

  for (int g = 0; g < NG; ++g) {
    float* Eb = sm + ((g & 1) ? E1_OFF : E0_OFF);
    if (g + 1 < NG) {                   // double-buffered DMA pipeline
      async_tile_load(Eg + (size_t)(g + 1) * GR * D_,
                      sm + (((g + 1) & 1) ? E1_OFF : E0_OFF));
      asm volatile("s_wait_asynccnt 16" ::: "memory");  // group g landed
    } else {
      asm volatile("s_wait_asynccnt 0" ::: "memory");
    }
    __syncthreads();

    // ---- S = codebook(32x256) . E_tile(16x256)^T  via 64 x wmma k=4 ----
    v8f s = vz;
    {
      const float* ap = cb + (mb * 16 + l16) * CBS + 2 * hi;  // A: M=class,K=d
      const float* bp = Eb + (t * 16 + l16) * ES + 2 * hi;    // B: K=d,N=n
#pragma unroll 8
      for (int k0 = 0; k0 < D_; k0 += 4) {
        v2f a  = *(const v2f*)(ap + k0);
        v2f bb = *(const v2f*)(bp + k0);
        s = wmma_f32_k4(a, bb, s);
      }
    }

    // ---- per-class max over this wave's 16 columns (ds_swizzle xor-reduce) ----
#pragma unroll
    for (int r = 0; r < 8; ++r) {
      float v = half_reduce_max(s[r]);
      if (l16 == 0) tmax[(mb * 16 + r + 8 * hi) * 4 + t] = v;
    }
    __syncthreads();
    if (wv == 0) {                       // online-softmax bookkeeping (32 classes)
      float mo = m_s[lane];
      float ml = fmaxf(fmaxf(tmax[lane*4+0], tmax[lane*4+1]),
                       fmaxf(tmax[lane*4+2], tmax[lane*4+3]));
      float mn = fmaxf(mo, ml);
      mn_s[lane] = mn;
      sc_s[lane] = __expf(mo - mn);
    }
    __syncthreads();

    // ---- P = exp(S - m_new) -> LDS; partial row sums ----
#pragma unroll
    for (int r = 0; r < 8; ++r) {
      const int c = mb * 16 + r + 8 * hi;
      float p = __expf(s[r] - mn_s[c]);
      Pm[c * PS + t * 16 + l16] = p;
      float su = half_reduce_sum(p);
      if (l16 == 0) tsum[c * 4 + t] = su;
    }
    __syncthreads();
    if (wv == 0) {
      l_s[lane] = l_s[lane] * sc_s[lane] +
                  tsum[lane*4+0] + tsum[lane*4+1] + tsum[lane*4+2] + tsum[lane*4+3];
      m_s[lane] = mn_s[lane];
    }

    // ---- rescale running acc, then acc += P(32x64) . E(64 x Dslice) ----
#pragma unroll
    for (int r = 0; r < 8; ++r) {
      const float f0 = sc_s[r + 8 * hi];
      const float f1 = sc_s[16 + r + 8 * hi];
      acc00[r] *= f0; acc01[r] *= f0;
      acc10[r] *= f1; acc11[r] *= f1;
    }
    {
      const int d0 = wv * 32;                       // this wave's D-slice
      const float* a0p = Pm + l16 * PS + 2 * hi;    // A frag, classes 0..15
      const float* a1p = Pm + (16 + l16) * PS + 2 * hi;  // classes 16..31
#pragma unroll 4
      for (int k0 = 0; k0 < GR; k0 += 4) {
        v2f a0 = *(const v2f*)(a0p + k0);
        v2f a1 = *(const v2f*)(a1p + k0);
        const float* e0 = Eb + (k0 + 2 * hi) * ES;
        v2f b0, b1;
        b0[0] = e0[d0 + l16];          b0[1] = e0[ES + d0 + l16];
        b1[0] = e0[d0 + 16 + l16];     b1[1] = e0[ES + 16 + d0 + l16];
        acc00 = wmma_f32_k4(a0, b0, acc00);
        acc10 = wmma_f32_k4(a1, b0, acc10);
        acc01 = wmma_f32_k4(a0, b1, acc01);
        acc11 = wmma_f32_k4(a1, b1, acc11);
      }
    }
    __syncthreads();   // protect E/P buffers before next group's DMA/writes
  }

  // ---- write unnormalized partial acc + (m,l) for this (b, chunk) ----
  {
    const int d0 = wv * 32;
    const size_t ob = (size_t)bq * CLS * D_;
#pragma unroll
    for (int r = 0; r < 8; ++r) {
      const int c0 = r + 8 * hi;           // < 16, always valid
      pAcc[ob + (size_t)c0 * D_ + d0 + l16]      = acc00[r];
      pAcc[ob + (size_t)c0 * D_ + d0 + 16 + l16] = acc01[r];
      const int c1 = 16 + r + 8 * hi;      // only 16..19 are real classes
      if (c1 < CLS) {
        pAcc[ob + (size_t)c1 * D_ + d0 + l16]      = acc10[r];
        pAcc[ob + (size_t)c1 * D_ + d0 + 16 + l16] = acc11[r];
      }
    }
    if (wv == 0 && lane < CLS) {
      pM[(size_t)bq * CLS + lane] = m_s[lane];
      pL[(size_t)bq * CLS + lane] = l_s[lane];
    }
  }
}

// Merge the CH chunk-partials per (b,c):  out = sum_q e^{m_q-M} acc_q / sum_q e^{m_q-M} l_q
__global__ __launch_bounds__(256, 4)
void combine(const float* __restrict__ pAcc, const float* __restrict__ pM,
             const float* __restrict__ pL, float* __restrict__ out) {
  const int bc = blockIdx.x;            // b*CLS + c
  const int b  = bc / CLS;
  const int c  = bc % CLS;
  const int d  = threadIdx.x;           // 256 = D_
  float mq[CH];
  float M = -3.0e38f;
#pragma unroll
  for (int q2 = 0; q2 < CH; ++q2) {
    mq[q2] = pM[((size_t)(b * CH + q2)) * CLS + c];
    M = fmaxf(M, mq[q2]);
  }
  float num = 0.f, den = 0.f;
#pragma unroll
  for (int q2 = 0; q2 < CH; ++q2) {
    const float wq = __expf(mq[q2] - M);
    num += wq * pAcc[(((size_t)(b * CH + q2)) * CLS + c) * D_ + d];
    den += wq * pL[((size_t)(b * CH + q2)) * CLS + c];
  }
  out[(size_t)bc * D_ + d] = num / den;
}

extern "C" void kernel_launch(void* const* d_in, const int* in_sizes, int n_in,
                              void* d_out, int out_size, void* d_ws, size_t ws_size,
                              hipStream_t stream) {
  const float* pe = (const float*)d_in[0];   // (64,4096,256) f32
  const float* cb = (const float*)d_in[1];   // (20,256) f32
  float* out  = (float*)d_out;               // (64,20,256) f32
  float* pAcc = (float*)d_ws;                               // 512*20*256 f32
  float* pM   = pAcc + (size_t)B_ * CH * CLS * D_;          // 512*20 f32
  float* pL   = pM + (size_t)B_ * CH * CLS;                 // 512*20 f32

  const size_t shmem = (size_t)LDS_FLOATS * sizeof(float);  // ~172.5 KB
  flash_partial<<<dim3(B_ * CH), dim3(256), shmem, stream>>>(pe, cb, pAcc, pM, pL);
  combine<<<dim3(B_ * CLS), dim3(256), 0, stream>>>(pAcc, pM, pL, out);
  (void)in_sizes; (void)n_in; (void)out_size; (void)ws_size;
}